// GCNNet_88467736363197
// MI455X (gfx1250) — compile-verified
//
#include <hip/hip_runtime.h>
#include <hip/hip_bf16.h>

typedef __attribute__((ext_vector_type(2))) float v2f;
typedef __attribute__((ext_vector_type(8))) float v8f;

#define NNODES 100000
#define NEDGES 800000
#define NB     2048
#define FXD    78

#define WMMA_F32(a, b, c) \
    __builtin_amdgcn_wmma_f32_16x16x4_f32(false, (a), false, (b), (short)0, (c), false, false)

// ---------------------------------------------------------------------------
// WMMA fp32 GEMM: C[M,N] = act(A[M,K] @ B + bias)
// B is [K,N] (TRANSB=0) or [N,K] (TRANSB=1), row-major, given leading dims.
// Each wave owns a 32x32 C tile: 4 accumulators, 2 A-frags + 2 B-frags per
// K-step of 4 -> 4 v_wmma_f32_16x16x4_f32 per step, unrolled x2.
// Interior tiles take a branch-free path; edge tiles use clamped addresses +
// mask multiplies (no per-lane EXEC divergence in the K loop).
// ---------------------------------------------------------------------------
__device__ __forceinline__ void store_tile(const v8f& acc, float* C, int ldc,
                                           int mbase, int ncol, int M, int N,
                                           float bv, int relu) {
    if (ncol >= N) return;
    #pragma unroll
    for (int r = 0; r < 8; ++r) {
        const int cm = mbase + r;
        if (cm < M) {
            float v = acc[r] + bv;
            if (relu) v = fmaxf(v, 0.0f);
            C[(long long)cm * ldc + ncol] = v;
        }
    }
}

template <int TRANSB>
__global__ void k_gemm_wmma(const float* __restrict__ A, int lda,
                            const float* __restrict__ B, int ldb,
                            const float* __restrict__ bias,
                            float* __restrict__ C, int ldc,
                            int M, int N, int K, int relu)
{
    const int lane = threadIdx.x & 31;
    const int wave = threadIdx.x >> 5;
    const int tilesN = (N + 31) >> 5;
    const int tilesM = (M + 31) >> 5;
    long long tile = (long long)blockIdx.x * (blockDim.x >> 5) + wave;
    if (tile >= (long long)tilesM * tilesN) return;  // uniform per wave
    const int tm = (int)(tile / tilesN) * 32;
    const int tn = (int)(tile % tilesN) * 32;

    const int l15  = lane & 15;
    const int koff = (lane >> 4) * 2;       // K sub-offset for A/B fragments

    const int m0 = tm + l15,      m1 = tm + 16 + l15;
    const int n0 = tn + l15,      n1 = tn + 16 + l15;

    v8f acc00 = {}, acc01 = {}, acc10 = {}, acc11 = {};

    const bool interior = (tm + 32 <= M) && (tn + 32 <= N);
    if (interior) {
        const float* A0 = A + (long long)m0 * lda;
        const float* A1 = A + (long long)m1 * lda;
        const float* Bp0 = TRANSB ? B + (long long)n0 * ldb : B + n0;
        const float* Bp1 = TRANSB ? B + (long long)n1 * ldb : B + n1;
        int k0 = 0;
        #pragma unroll 2
        for (; k0 + 4 <= K; k0 += 4) {
            const int ka = k0 + koff;
            __builtin_prefetch(&A0[k0 + 128], 0, 0);
            v2f a0 = {A0[ka], A0[ka + 1]};
            v2f a1 = {A1[ka], A1[ka + 1]};
            v2f b0, b1;
            if (TRANSB) {
                b0 = v2f{Bp0[ka], Bp0[ka + 1]};
                b1 = v2f{Bp1[ka], Bp1[ka + 1]};
            } else {
                b0 = v2f{Bp0[(long long)ka * ldb], Bp0[(long long)(ka + 1) * ldb]};
                b1 = v2f{Bp1[(long long)ka * ldb], Bp1[(long long)(ka + 1) * ldb]};
            }
            acc00 = WMMA_F32(a0, b0, acc00);
            acc01 = WMMA_F32(a0, b1, acc01);
            acc10 = WMMA_F32(a1, b0, acc10);
            acc11 = WMMA_F32(a1, b1, acc11);
        }
        if (k0 < K) {  // K % 4 tail: clamped addresses, cndmask zeros
            const int ka = k0 + koff, kb = ka + 1;
            const int kac = (ka < K) ? ka : 0;
            const int kbc = (kb < K) ? kb : 0;
            v2f a0 = {(ka < K) ? A0[kac] : 0.0f, (kb < K) ? A0[kbc] : 0.0f};
            v2f a1 = {(ka < K) ? A1[kac] : 0.0f, (kb < K) ? A1[kbc] : 0.0f};
            v2f b0, b1;
            if (TRANSB) {
                b0 = v2f{(ka < K) ? Bp0[kac] : 0.0f, (kb < K) ? Bp0[kbc] : 0.0f};
                b1 = v2f{(ka < K) ? Bp1[kac] : 0.0f, (kb < K) ? Bp1[kbc] : 0.0f};
            } else {
                b0 = v2f{(ka < K) ? Bp0[(long long)kac * ldb] : 0.0f,
                         (kb < K) ? Bp0[(long long)kbc * ldb] : 0.0f};
                b1 = v2f{(ka < K) ? Bp1[(long long)kac * ldb] : 0.0f,
                         (kb < K) ? Bp1[(long long)kbc * ldb] : 0.0f};
            }
            acc00 = WMMA_F32(a0, b0, acc00);
            acc01 = WMMA_F32(a0, b1, acc01);
            acc10 = WMMA_F32(a1, b0, acc10);
            acc11 = WMMA_F32(a1, b1, acc11);
        }
    } else {
        // edge tiles: clamp row/col addresses, zero via mask multiply
        const float fm0 = (m0 < M) ? 1.0f : 0.0f;
        const float fm1 = (m1 < M) ? 1.0f : 0.0f;
        const float fn0 = (n0 < N) ? 1.0f : 0.0f;
        const float fn1 = (n1 < N) ? 1.0f : 0.0f;
        const float* A0 = A + (long long)((m0 < M) ? m0 : 0) * lda;
        const float* A1 = A + (long long)((m1 < M) ? m1 : 0) * lda;
        const int n0c = (n0 < N) ? n0 : 0;
        const int n1c = (n1 < N) ? n1 : 0;
        const float* Bp0 = TRANSB ? B + (long long)n0c * ldb : B + n0c;
        const float* Bp1 = TRANSB ? B + (long long)n1c * ldb : B + n1c;
        for (int k0 = 0; k0 < K; k0 += 4) {
            const int ka = k0 + koff, kb = ka + 1;
            const int kac = (ka < K) ? ka : 0;
            const int kbc = (kb < K) ? kb : 0;
            const float fka = (ka < K) ? 1.0f : 0.0f;
            const float fkb = (kb < K) ? 1.0f : 0.0f;
            v2f a0 = {A0[kac] * (fm0 * fka), A0[kbc] * (fm0 * fkb)};
            v2f a1 = {A1[kac] * (fm1 * fka), A1[kbc] * (fm1 * fkb)};
            v2f b0, b1;
            if (TRANSB) {
                b0 = v2f{Bp0[kac] * (fn0 * fka), Bp0[kbc] * (fn0 * fkb)};
                b1 = v2f{Bp1[kac] * (fn1 * fka), Bp1[kbc] * (fn1 * fkb)};
            } else {
                b0 = v2f{Bp0[(long long)kac * ldb] * (fn0 * fka),
                         Bp0[(long long)kbc * ldb] * (fn0 * fkb)};
                b1 = v2f{Bp1[(long long)kac * ldb] * (fn1 * fka),
                         Bp1[(long long)kbc * ldb] * (fn1 * fkb)};
            }
            acc00 = WMMA_F32(a0, b0, acc00);
            acc01 = WMMA_F32(a0, b1, acc01);
            acc10 = WMMA_F32(a1, b0, acc10);
            acc11 = WMMA_F32(a1, b1, acc11);
        }
    }

    // epilogue (divergence here is fine: all WMMAs are done)
    const int mb0 = tm + ((lane >> 4) ? 8 : 0);
    const int mb1 = mb0 + 16;
    const float bv0 = (bias && n0 < N) ? bias[n0] : 0.0f;
    const float bv1 = (bias && n1 < N) ? bias[n1] : 0.0f;
    store_tile(acc00, C, ldc, mb0, n0, M, N, bv0, relu);
    store_tile(acc01, C, ldc, mb0, n1, M, N, bv1, relu);
    store_tile(acc10, C, ldc, mb1, n0, M, N, bv0, relu);
    store_tile(acc11, C, ldc, mb1, n1, M, N, bv1, relu);
}

// ---------------------------------------------------------------------------
// GCN helpers (wave-per-edge / wave-per-node for coalesced feature rows)
// ---------------------------------------------------------------------------
__global__ void k_deg_init(float* dinv, int n) {
    int i = blockIdx.x * blockDim.x + threadIdx.x;
    if (i < n) dinv[i] = 1.0f;   // self-loop
}
__global__ void k_deg_count(const int* __restrict__ dst, float* dinv, int e) {
    int i = blockIdx.x * blockDim.x + threadIdx.x;
    if (i < e) atomicAdd(&dinv[dst[i]], 1.0f);
}
__global__ void k_deg_final(float* dinv, int n) {
    int i = blockIdx.x * blockDim.x + threadIdx.x;
    if (i < n) dinv[i] = rsqrtf(dinv[i]);
}
// out[n,f] = dinv[n]^2 * in[n,f]   (self-loop term, initializes out)
__global__ void k_agg_self(const float* __restrict__ in, const float* __restrict__ dinv,
                           float* __restrict__ out, int F, int Nn) {
    int nid = blockIdx.x * (blockDim.x >> 5) + (threadIdx.x >> 5);
    if (nid >= Nn) return;
    const int lane = threadIdx.x & 31;
    const float w = dinv[nid] * dinv[nid];
    const float* pi = in  + (long long)nid * F;
    float*       po = out + (long long)nid * F;
    for (int f = lane; f < F; f += 32) po[f] = w * pi[f];
}
// out[dst,:] += dinv[src]*dinv[dst]*in[src,:]
__global__ void k_agg_edges(const int* __restrict__ src, const int* __restrict__ dst,
                            const float* __restrict__ dinv,
                            const float* __restrict__ in, float* __restrict__ out,
                            int F, int E) {
    int e = blockIdx.x * (blockDim.x >> 5) + (threadIdx.x >> 5);
    if (e >= E) return;
    const int lane = threadIdx.x & 31;
    const int s = src[e], d = dst[e];
    const float w = dinv[s] * dinv[d];
    const float* pi = in  + (long long)s * F;
    float*       po = out + (long long)d * F;
    for (int f = lane; f < F; f += 32) atomicAdd(&po[f], w * pi[f]);
}
__global__ void k_bias_relu(float* __restrict__ h, const float* __restrict__ b,
                            int F, int Nn) {
    int nid = blockIdx.x * (blockDim.x >> 5) + (threadIdx.x >> 5);
    if (nid >= Nn) return;
    const int lane = threadIdx.x & 31;
    float* p = h + (long long)nid * F;
    for (int f = lane; f < F; f += 32) p[f] = fmaxf(p[f] + b[f], 0.0f);
}
__global__ void k_zero(float* p, long long total) {
    long long t = (long long)blockIdx.x * blockDim.x + threadIdx.x;
    if (t < total) p[t] = 0.0f;
}
// segment max pool; h >= 0 post-ReLU so uint bit-pattern atomicMax is exact,
// and the 0-init reproduces the isfinite guard for empty graphs.
__global__ void k_pool_max(const float* __restrict__ h, const int* __restrict__ batch,
                           float* __restrict__ g, int F, int Nn) {
    int nid = blockIdx.x * (blockDim.x >> 5) + (threadIdx.x >> 5);
    if (nid >= Nn) return;
    const int lane = threadIdx.x & 31;
    const float* pi = h + (long long)nid * F;
    unsigned int* po = (unsigned int*)(g + (long long)batch[nid] * F);
    for (int f = lane; f < F; f += 32) atomicMax(&po[f], __float_as_uint(pi[f]));
}
// row softmax with pre-scale
__global__ void k_softmax(float* __restrict__ S, int cols, float scale) {
    float* p = S + (long long)blockIdx.x * cols;
    __shared__ float red[256];
    float m = -1e30f;
    for (int c = threadIdx.x; c < cols; c += blockDim.x) m = fmaxf(m, p[c] * scale);
    red[threadIdx.x] = m; __syncthreads();
    for (int s = blockDim.x >> 1; s > 0; s >>= 1) {
        if ((int)threadIdx.x < s) red[threadIdx.x] = fmaxf(red[threadIdx.x], red[threadIdx.x + s]);
        __syncthreads();
    }
    m = red[0]; __syncthreads();
    float sum = 0.0f;
    for (int c = threadIdx.x; c < cols; c += blockDim.x) {
        float e = __expf(p[c] * scale - m);
        p[c] = e; sum += e;
    }
    red[threadIdx.x] = sum; __syncthreads();
    for (int s = blockDim.x >> 1; s > 0; s >>= 1) {
        if ((int)threadIdx.x < s) red[threadIdx.x] += red[threadIdx.x + s];
        __syncthreads();
    }
    const float inv = 1.0f / red[0];
    for (int c = threadIdx.x; c < cols; c += blockDim.x) p[c] *= inv;
}

// ---------------------------------------------------------------------------
// Host side
// ---------------------------------------------------------------------------
static inline void gemm(const float* A, int lda, const float* B, int ldb,
                        const float* bias, float* C, int ldc,
                        int M, int N, int K, int transB, int relu, hipStream_t s) {
    long long tiles = (long long)((M + 31) / 32) * ((N + 31) / 32);
    unsigned blocks = (unsigned)((tiles + 7) / 8);   // 8 waves (256 thr) per block
    if (transB)
        k_gemm_wmma<1><<<blocks, 256, 0, s>>>(A, lda, B, ldb, bias, C, ldc, M, N, K, relu);
    else
        k_gemm_wmma<0><<<blocks, 256, 0, s>>>(A, lda, B, ldb, bias, C, ldc, M, N, K, relu);
}
static inline unsigned gsz(long long total) { return (unsigned)((total + 255) / 256); }
static inline unsigned gw(long long items) { return (unsigned)((items + 7) / 8); } // 8 waves/block

extern "C" void kernel_launch(void* const* d_in, const int* in_sizes, int n_in,
                              void* d_out, int out_size, void* d_ws, size_t ws_size,
                              hipStream_t stream) {
    (void)in_sizes; (void)n_in; (void)out_size; (void)ws_size;
    const float* x        = (const float*)d_in[0];
    const int*   esrc     = (const int*)  d_in[1];
    const int*   edst     = (const int*)  d_in[2];
    const int*   batch    = (const int*)  d_in[3];
    const float* proseq   = (const float*)d_in[4];
    const float* smiles   = (const float*)d_in[5];
    const float* gW1 = (const float*)d_in[6],  *gb1 = (const float*)d_in[7];
    const float* gW2 = (const float*)d_in[8],  *gb2 = (const float*)d_in[9];
    const float* gW3 = (const float*)d_in[10], *gb3 = (const float*)d_in[11];
    const float* dfc1W = (const float*)d_in[12], *dfc1b = (const float*)d_in[13];
    const float* dfc2W = (const float*)d_in[14], *dfc2b = (const float*)d_in[15];
    const float* pfc1W = (const float*)d_in[16], *pfc1b = (const float*)d_in[17];
    const float* pfc2W = (const float*)d_in[18], *pfc2b = (const float*)d_in[19];
    const float* efc1W = (const float*)d_in[20], *efc1b = (const float*)d_in[21];
    const float* efc2W = (const float*)d_in[22], *efc2b = (const float*)d_in[23];
    const float* WQ  = (const float*)d_in[24];
    const float* WK  = (const float*)d_in[25];
    const float* WV  = (const float*)d_in[26];
    const float* Wfc = (const float*)d_in[27];
    const float* fc1W = (const float*)d_in[28], *fc1b = (const float*)d_in[29];
    const float* fc2W = (const float*)d_in[30], *fc2b = (const float*)d_in[31];
    const float* outW = (const float*)d_in[32], *outb = (const float*)d_in[33];
    float* out = (float*)d_out;

    // workspace arena (floats)
    float* dinv = (float*)d_ws;                          // NNODES
    float* bufA = dinv + NNODES;                         // NNODES*312
    float* bufB = bufA + (size_t)NNODES * 312;           // NNODES*312
    float* g    = bufB + (size_t)NNODES * 312;           // NB*312
    // batch phase reuses bufA (28.3M floats < 31.2M available)
    float* tmp    = bufA;                                // NB*2048
    float* xc2    = tmp    + (size_t)NB * 2048;          // NB*3072
    float* Q      = xc2    + (size_t)NB * 3072;          // NB*1536
    float* Km     = Q      + (size_t)NB * 1536;          // NB*1536
    float* V      = Km     + (size_t)NB * 1536;          // NB*1536
    float* scores = V      + (size_t)NB * 1536;          // NB*2048
    float* ctx    = scores + (size_t)NB * 2048;          // NB*1536
    float* f2     = ctx    + (size_t)NB * 1536;          // NB*512

    // ---- degree / normalization ----
    k_deg_init <<<gsz(NNODES), 256, 0, stream>>>(dinv, NNODES);
    k_deg_count<<<gsz(NEDGES), 256, 0, stream>>>(edst, dinv, NEDGES);
    k_deg_final<<<gsz(NNODES), 256, 0, stream>>>(dinv, NNODES);

    // ---- 3 GCN layers: XW (WMMA) -> self+edge aggregation -> bias+ReLU ----
    const float* hin = x;
    const float* Ws[3] = {gW1, gW2, gW3};
    const float* bs[3] = {gb1, gb2, gb3};
    const int    Kd[3] = {FXD, FXD, 2 * FXD};
    const int    Fd[3] = {FXD, 2 * FXD, 4 * FXD};
    for (int l = 0; l < 3; ++l) {
        const int F = Fd[l];
        gemm(hin, Kd[l], Ws[l], F, nullptr, bufA, F, NNODES, F, Kd[l], 0, 0, stream);
        k_agg_self <<<gw(NNODES), 256, 0, stream>>>(bufA, dinv, bufB, F, NNODES);
        k_agg_edges<<<gw(NEDGES), 256, 0, stream>>>(esrc, edst, dinv, bufA, bufB, F, NEDGES);
        k_bias_relu<<<gw(NNODES), 256, 0, stream>>>(bufB, bs[l], F, NNODES);
        hin = bufB;
    }

    // ---- global max pool ----
    k_zero    <<<gsz((long long)NB * 312), 256, 0, stream>>>(g, (long long)NB * 312);
    k_pool_max<<<gw(NNODES), 256, 0, stream>>>(bufB, batch, g, 312, NNODES);

    // ---- three MLP branches, written into xc2[:, 0:1536] (free concat) ----
    gemm(g, 312, dfc1W, 1024, dfc1b, tmp, 1024, NB, 1024, 312, 0, 1, stream);
    gemm(tmp, 1024, dfc2W, 512, dfc2b, xc2 + 0, 3072, NB, 512, 1024, 0, 1, stream);

    gemm(proseq, 4096, pfc1W, 2048, pfc1b, tmp, 2048, NB, 2048, 4096, 0, 1, stream);
    gemm(tmp, 2048, pfc2W, 512, pfc2b, xc2 + 512, 3072, NB, 512, 2048, 0, 1, stream);

    gemm(smiles, 512, efc1W, 1024, efc1b, tmp, 1024, NB, 1024, 512, 0, 1, stream);
    gemm(tmp, 1024, efc2W, 512, efc2b, xc2 + 1024, 3072, NB, 512, 1024, 0, 1, stream);

    // ---- MHA over xc = xc2[:, 0:1536], 4 heads, dk = 384 ----
    gemm(xc2, 3072, WQ, 1536, nullptr, Q,  1536, NB, 1536, 1536, 0, 0, stream);
    gemm(xc2, 3072, WK, 1536, nullptr, Km, 1536, NB, 1536, 1536, 0, 0, stream);
    gemm(xc2, 3072, WV, 1536, nullptr, V,  1536, NB, 1536, 1536, 0, 0, stream);
    const float scale = 0.05103103630798287f;  // 1/sqrt(384)
    for (int h = 0; h < 4; ++h) {
        const int o = h * 384;
        gemm(Q + o, 1536, Km + o, 1536, nullptr, scores, 2048, NB, NB, 384, 1, 0, stream);
        k_softmax<<<NB, 256, 0, stream>>>(scores, NB, scale);
        gemm(scores, 2048, V + o, 1536, nullptr, ctx + o, 1536, NB, 384, NB, 0, 0, stream);
    }
    gemm(ctx, 1536, Wfc, 1536, nullptr, xc2 + 1536, 3072, NB, 1536, 1536, 0, 0, stream);

    // ---- head: fc1 -> fc2 -> out ----
    gemm(xc2, 3072, fc1W, 1024, fc1b, tmp, 1024, NB, 1024, 3072, 0, 1, stream);
    gemm(tmp, 1024, fc2W, 512, fc2b, f2, 512, NB, 512, 1024, 0, 1, stream);
    gemm(f2, 512, outW, 2, outb, out, 2, NB, 2, 512, 0, 0, stream);
}